// _HomoBackbone_31293131719110
// MI455X (gfx1250) — compile-verified
//
#include <hip/hip_runtime.h>

typedef __attribute__((ext_vector_type(2))) float v2f;
typedef __attribute__((ext_vector_type(8))) float v8f;

#define HID 64
#define LDS_STRIDE 66   // even (8B-aligned b64 LDS loads) and 66 % 64 = 2 -> conflict-free column reads

// ---------------- utility kernels ----------------

__global__ __launch_bounds__(256) void k_zero(float* __restrict__ p, long n) {
  long i = (long)blockIdx.x * blockDim.x + threadIdx.x;
  if (i < n) p[i] = 0.0f;
}

__global__ __launch_bounds__(256) void k_degree(const long long* __restrict__ col,
                                                float* __restrict__ deg, int E) {
  int e = blockIdx.x * blockDim.x + threadIdx.x;
  if (e < E) atomicAdd(&deg[col[e]], 1.0f);
}

__global__ __launch_bounds__(256) void k_dinv(float* __restrict__ deg, int N) {
  int i = blockIdx.x * blockDim.x + threadIdx.x;
  if (i < N) {
    float d = deg[i];
    deg[i] = (d > 0.0f) ? rsqrtf(d) : 0.0f;   // deg integral counts; max(d,1) redundant for d>0
  }
}

__global__ __launch_bounds__(256) void k_norm(const long long* __restrict__ ei,
                                              const float* __restrict__ dinv,
                                              float* __restrict__ norm, int E) {
  int e = blockIdx.x * blockDim.x + threadIdx.x;
  if (e < E) {
    long long r = ei[e];
    long long c = ei[(long)E + e];
    norm[e] = dinv[r] * dinv[c];
  }
}

// ---------------- WMMA GEMM: H = X @ W  (fp32, 16x16x4) ----------------
// One wave -> 16 rows x 64 cols. W staged transposed into LDS so a B fragment
// (two consecutive K for one column) is a contiguous 8-byte LDS load.

__global__ __launch_bounds__(256) void k_gemm_wmma(const float* __restrict__ X,
                                                   const float* __restrict__ W,
                                                   float* __restrict__ H, int nStrips) {
  __shared__ float WT[HID * LDS_STRIDE];   // WT[col][k] = W[k][col]
  for (int i = threadIdx.x; i < HID * HID; i += 256) {
    int k = i >> 6, c = i & 63;
    WT[c * LDS_STRIDE + k] = W[i];
  }
  __syncthreads();

  const int wave = threadIdx.x >> 5;
  const int lane = threadIdx.x & 31;
  const int strip = blockIdx.x * 8 + wave;
  if (strip >= nStrips) return;            // wave-uniform: EXEC stays all-ones for WMMA

  // A layout (16x4 f32): lanes 0-15 -> M=lane, K={0,1}; lanes 16-31 -> M=lane-16, K={2,3}
  const int mrow = (strip << 4) + (lane & 15);
  const int koff = (lane >> 4) << 1;       // 0 or 2
  const int ncol = lane & 15;              // B/D column within a 16-wide tile

  const float* xrow = X + (long)mrow * HID;

  v8f acc0 = {}, acc1 = {}, acc2 = {}, acc3 = {};
  #pragma unroll
  for (int kk = 0; kk < HID; kk += 4) {
    v2f a = *(const v2f*)(xrow + kk + koff);
    v2f b0 = *(const v2f*)(&WT[(ncol +  0) * LDS_STRIDE + kk + koff]);
    v2f b1 = *(const v2f*)(&WT[(ncol + 16) * LDS_STRIDE + kk + koff]);
    v2f b2 = *(const v2f*)(&WT[(ncol + 32) * LDS_STRIDE + kk + koff]);
    v2f b3 = *(const v2f*)(&WT[(ncol + 48) * LDS_STRIDE + kk + koff]);
    acc0 = __builtin_amdgcn_wmma_f32_16x16x4_f32(false, a, false, b0, (short)0, acc0, false, false);
    acc1 = __builtin_amdgcn_wmma_f32_16x16x4_f32(false, a, false, b1, (short)0, acc1, false, false);
    acc2 = __builtin_amdgcn_wmma_f32_16x16x4_f32(false, a, false, b2, (short)0, acc2, false, false);
    acc3 = __builtin_amdgcn_wmma_f32_16x16x4_f32(false, a, false, b3, (short)0, acc3, false, false);
  }

  // D layout: VGPR v -> row = strip*16 + v + (lane>=16 ? 8 : 0), col = tile*16 + (lane&15)
  const int rbase = (strip << 4) + ((lane >> 4) << 3);
  float* hp = H + (long)rbase * HID + ncol;
  #pragma unroll
  for (int v = 0; v < 8; ++v) {
    hp[(long)v * HID +  0] = acc0[v];
    hp[(long)v * HID + 16] = acc1[v];
    hp[(long)v * HID + 32] = acc2[v];
    hp[(long)v * HID + 48] = acc3[v];
  }
}

// scalar tail for N % 16 rows (not taken for N=100000, kept for generality)
__global__ __launch_bounds__(256) void k_gemm_tail(const float* __restrict__ X,
                                                   const float* __restrict__ W,
                                                   float* __restrict__ H,
                                                   int rowStart, int N) {
  int t = blockIdx.x * blockDim.x + threadIdx.x;
  int row = rowStart + t / HID;
  int c = t % HID;
  if (row < N) {
    float s = 0.0f;
    for (int k = 0; k < HID; ++k) s += X[(long)row * HID + k] * W[k * HID + c];
    H[(long)row * HID + c] = s;
  }
}

// ---------------- edge scatter: agg[col] += h[row] * norm ----------------
// 16 lanes per edge, float4 chunk each -> fully coalesced 256B row traffic.

__global__ __launch_bounds__(256) void k_scatter(const float* __restrict__ h,
                                                 const long long* __restrict__ ei,
                                                 const float* __restrict__ norm,
                                                 float* __restrict__ agg, int E) {
  long tid = (long)blockIdx.x * blockDim.x + threadIdx.x;
  long e = tid >> 4;
  int c = (int)(tid & 15) << 2;
  if (e < E) {
    long long r = ei[e];
    long long cl = ei[(long)E + e];
    float nv = norm[e];
    float4 v = *(const float4*)(h + (long)r * HID + c);
    float* dst = agg + (long)cl * HID + c;
    atomicAdd(dst + 0, v.x * nv);
    atomicAdd(dst + 1, v.y * nv);
    atomicAdd(dst + 2, v.z * nv);
    atomicAdd(dst + 3, v.w * nv);
  }
}

__global__ __launch_bounds__(256) void k_bias_relu(const float* __restrict__ agg,
                                                   const float* __restrict__ b,
                                                   float* __restrict__ out, long n) {
  long i = (long)blockIdx.x * blockDim.x + threadIdx.x;
  if (i < n) {
    float v = agg[i] + b[i & (HID - 1)];
    out[i] = v > 0.0f ? v : 0.0f;
  }
}

// ---------------- driver ----------------

extern "C" void kernel_launch(void* const* d_in, const int* in_sizes, int n_in,
                              void* d_out, int out_size, void* d_ws, size_t ws_size,
                              hipStream_t stream) {
  (void)n_in; (void)out_size; (void)ws_size;

  const float*     x0 = (const float*)d_in[0];
  const long long* ei = (const long long*)d_in[1];
  const float* Wm[3] = {(const float*)d_in[2], (const float*)d_in[4], (const float*)d_in[6]};
  const float* Bm[3] = {(const float*)d_in[3], (const float*)d_in[5], (const float*)d_in[7]};

  const int  N  = in_sizes[0] / HID;
  const int  E  = in_sizes[1] / 2;
  const long NH = (long)N * HID;

  float* deg  = (float*)d_ws;      // N
  float* norm = deg  + N;          // E
  float* hbuf = norm + E;          // N*H
  float* aggb = hbuf + NH;         // N*H
  float* xbuf = aggb + NH;         // N*H

  dim3 blk(256);
  auto g1 = [](long n) { return dim3((unsigned)((n + 255) / 256)); };

  // gcn_norm: deg (scatter) -> deg^-1/2 -> per-edge norm
  k_zero  <<<g1(N), blk, 0, stream>>>(deg, N);
  k_degree<<<g1(E), blk, 0, stream>>>(ei + E, deg, E);
  k_dinv  <<<g1(N), blk, 0, stream>>>(deg, N);
  k_norm  <<<g1(E), blk, 0, stream>>>(ei, deg, norm, E);

  const int nStrips = N / 16;
  const int rem = N - nStrips * 16;
  dim3 gemmGrid((unsigned)((nStrips + 7) / 8));

  const float* cur = x0;
  for (int l = 0; l < 3; ++l) {
    k_gemm_wmma<<<gemmGrid, blk, 0, stream>>>(cur, Wm[l], hbuf, nStrips);
    if (rem)
      k_gemm_tail<<<g1((long)rem * HID), blk, 0, stream>>>(cur, Wm[l], hbuf, nStrips * 16, N);
    k_zero   <<<g1(NH), blk, 0, stream>>>(aggb, NH);
    k_scatter<<<g1((long)E * 16), blk, 0, stream>>>(hbuf, ei, norm, aggb, E);
    float* dst = (l == 2) ? (float*)d_out : xbuf;
    k_bias_relu<<<g1(NH), blk, 0, stream>>>(aggb, Bm[l], dst, NH);
    cur = xbuf;
  }
}